// RandomPhongShader_29935922053316
// MI455X (gfx1250) — compile-verified
//
#include <hip/hip_runtime.h>
#include <cstdint>
#include <cstddef>

// ---------------------------------------------------------------------------
// RandomPhongShader for MI455X (gfx1250, wave32).
//
// VALU-bound (threefry2x32-20 RNG + erfinv); the ~109 MB of memory traffic
// (~4.7us at 23.3 TB/s) hides under compute. The 50 MB color tensor streams
// through the CDNA5 async global->LDS DMA path (GLOBAL_LOAD_ASYNC_TO_LDS_B128,
// ASYNCcnt) issued at kernel start and consumed after all RNG work behind
// s_wait_asynccnt.
// ---------------------------------------------------------------------------

#define AS(n) __attribute__((address_space(n)))
typedef int v4i __attribute__((ext_vector_type(4)));

#if defined(__HIP_DEVICE_COMPILE__) && \
    __has_builtin(__builtin_amdgcn_global_load_async_to_lds_b128)
#define HAVE_ASYNC_LDS 1
#else
#define HAVE_ASYNC_LDS 0
#endif

struct U2 { uint32_t a, b; };

__host__ __device__ constexpr uint32_t rotl32(uint32_t x, int d) {
  return (x << d) | (x >> (32 - d));
}

#define TF_ROUND(r) do { x0 += x1; x1 = rotl32(x1, (r)); x1 ^= x0; } while (0);

// Threefry-2x32, 20 rounds — exactly JAX's threefry2x32.
__host__ __device__ constexpr U2 tf2x32(uint32_t k0, uint32_t k1,
                                        uint32_t x0, uint32_t x1) {
  const uint32_t ks0 = k0, ks1 = k1, ks2 = k0 ^ k1 ^ 0x1BD11BDAu;
  x0 += ks0; x1 += ks1;
  TF_ROUND(13) TF_ROUND(15) TF_ROUND(26) TF_ROUND(6)
  x0 += ks1; x1 += ks2 + 1u;
  TF_ROUND(17) TF_ROUND(29) TF_ROUND(16) TF_ROUND(24)
  x0 += ks2; x1 += ks0 + 2u;
  TF_ROUND(13) TF_ROUND(15) TF_ROUND(26) TF_ROUND(6)
  x0 += ks0; x1 += ks1 + 3u;
  TF_ROUND(17) TF_ROUND(29) TF_ROUND(16) TF_ROUND(24)
  x0 += ks1; x1 += ks2 + 4u;
  TF_ROUND(13) TF_ROUND(15) TF_ROUND(26) TF_ROUND(6)
  x0 += ks2; x1 += ks0 + 5u;
  return U2{x0, x1};
}

// jax.random.key(1) -> (0,1); jax.random.split -> two threefry blocks,
// folded to literals at compile time.
constexpr U2 kSplitBlk0 = tf2x32(0u, 1u, 0u, 2u);
constexpr U2 kSplitBlk1 = tf2x32(0u, 1u, 1u, 3u);
constexpr uint32_t KH0 = kSplitBlk0.a, KH1 = kSplitBlk1.a;  // key for noise_h
constexpr uint32_t KA0 = kSplitBlk0.b, KA1 = kSplitBlk1.b;  // key for noise_a

// JAX uniform: f in [0,1) from mantissa bits; u = f*(hi-lo) + lo with
// lo = nextafterf(-1,0). (hi-lo) = 2-2^-24 rounds to exactly 2.0f in fp32,
// and f*2.0f is exact, so this fmaf reproduces XLA's mul+add bit-exactly.
__device__ __forceinline__ float bits_to_unif(uint32_t b) {
  float f = __uint_as_float((b >> 9) | 0x3f800000u) - 1.0f;  // [0,1)
  return fmaf(f, 2.0f, -0.99999994f);
}

// Giles single-precision erfinv (same polynomial XLA uses for erf_inv f32).
__device__ __forceinline__ float erfinv_f32(float x) {
  float w = -__logf(fmaf(-x, x, 1.0f));
  float p;
  if (w < 5.0f) {
    w = w - 2.5f;
    p =                2.81022636e-08f;
    p = fmaf(p, w,     3.43273939e-07f);
    p = fmaf(p, w,    -3.5233877e-06f);
    p = fmaf(p, w,    -4.39150654e-06f);
    p = fmaf(p, w,     0.00021858087f);
    p = fmaf(p, w,    -0.00125372503f);
    p = fmaf(p, w,    -0.00417768164f);
    p = fmaf(p, w,     0.246640727f);
    p = fmaf(p, w,     1.50140941f);
  } else {
    w = __fsqrt_rn(w) - 3.0f;
    p =               -0.000200214257f;
    p = fmaf(p, w,     0.000100950558f);
    p = fmaf(p, w,     0.00134934322f);
    p = fmaf(p, w,    -0.00367342844f);
    p = fmaf(p, w,     0.00573950773f);
    p = fmaf(p, w,    -0.0076224613f);
    p = fmaf(p, w,     0.00943887047f);
    p = fmaf(p, w,     1.00167406f);
    p = fmaf(p, w,     2.83297682f);
  }
  return p * x;
}

__device__ __forceinline__ float bits_to_normal(uint32_t b) {
  return 1.41421356237309505f * erfinv_f32(bits_to_unif(b));
}

#if HAVE_ASYNC_LDS
// Builtin signature (probed): (v4i AS(1)*, v4i AS(3)*, imm offset, imm cpol).
__device__ __forceinline__ AS(1) v4i* gbl_cast(const void* p) {
  return (AS(1) v4i*)(uintptr_t)p;
}
__device__ __forceinline__ AS(3) v4i* lds_cast(void* p) {
  // Generic LDS address: low 32 bits are the LDS byte offset (aperture rule).
  return (AS(3) v4i*)(uint32_t)(uint64_t)(uintptr_t)p;
}
#endif

// N=8, H=256, W=256, K=8
#define NPIX   524288u              // N*H*W
#define SH_STR 4194304u             // N*H*W*K   (noise_h per-sample stride)
#define SA_STR 4718592u             // N*H*W*9   (noise_a per-sample stride)

__global__ __launch_bounds__(256)
void rps_kernel(const float* __restrict__ colors,
                const float* __restrict__ dists,
                const float* __restrict__ zbuf,
                const int*   __restrict__ ptf,
                const float* __restrict__ bg,
                float* __restrict__ out) {
  const int tid = threadIdx.x;
  const uint32_t pix = blockIdx.x * 256u + (uint32_t)tid;

  const float* gcol = colors + (size_t)pix * 24u;

#if HAVE_ASYNC_LDS
  // Stage 24 color floats / pixel through LDS with the async DMA path so the
  // 50 MB color stream overlaps the RNG compute. Stride 28 floats = 112 B
  // keeps 16 B alignment for B128 and limits bank conflicts.
  __shared__ float lcol[256 * 28];
  float* mycol = &lcol[tid * 28];
#pragma unroll
  for (int j = 0; j < 6; ++j) {
    __builtin_amdgcn_global_load_async_to_lds_b128(
        gbl_cast(gcol + j * 4), lds_cast(mycol + j * 4), 0, 0);
  }
#endif

  const size_t kbase = (size_t)pix * 8u;
  const float4 d0 = *(const float4*)(dists + kbase);
  const float4 d1 = *(const float4*)(dists + kbase + 4);
  const int4   p0 = *(const int4*)(ptf + kbase);
  const int4   p1 = *(const int4*)(ptf + kbase + 4);
  const float4 z0 = *(const float4*)(zbuf + kbase);
  const float4 z1 = *(const float4*)(zbuf + kbase + 4);

  const float dk[8] = {d0.x, d0.y, d0.z, d0.w, d1.x, d1.y, d1.z, d1.w};
  const float zk[8] = {z0.x, z0.y, z0.z, z0.w, z1.x, z1.y, z1.z, z1.w};
  const float mk[8] = {p0.x >= 0 ? 1.f : 0.f, p0.y >= 0 ? 1.f : 0.f,
                       p0.z >= 0 ? 1.f : 0.f, p0.w >= 0 ? 1.f : 0.f,
                       p1.x >= 0 ? 1.f : 0.f, p1.y >= 0 ? 1.f : 0.f,
                       p1.z >= 0 ? 1.f : 0.f, p1.w >= 0 ? 1.f : 0.f};

  // ---- prob_map: heaviside via uniform threshold (u >= erf(d/(sigma*sqrt2)))
  // (monotone-equivalent to -d + sigma*sqrt2*erfinv(u) >= 0, saves 32 erfinv).
  // s=0/s=2 share one threefry block (x0/x1 outputs), likewise s=1/s=3.
  const uint32_t baseH = pix * 8u;
  int cntk[8];
  float prob[8];
#pragma unroll
  for (int k = 0; k < 8; ++k) {
    const float t = erff(dk[k] * 7.07106781186547524f);  // 1/(SIGMA*sqrt(2))
    const uint32_t c0 = baseH + (uint32_t)k;
    const U2 r02 = tf2x32(KH0, KH1, c0,          c0 + 2u * SH_STR);
    const U2 r13 = tf2x32(KH0, KH1, c0 + SH_STR, c0 + 3u * SH_STR);
    cntk[k] = (bits_to_unif(r02.a) >= t) + (bits_to_unif(r13.a) >= t) +
              (bits_to_unif(r02.b) >= t) + (bits_to_unif(r13.b) >= t);
    prob[k] = 0.25f * (float)cntk[k] * mk[k];   // exact in fp32
  }

  // ---- alpha, inverse depth, z_map
  float alpha = 1.0f;
  float zmax = 1e-10f;
  float zinv[8];
#pragma unroll
  for (int k = 0; k < 8; ++k) {
    alpha *= (1.0f - prob[k]);                       // exact products
    zinv[k] = ((100.0f - zk[k]) / 99.0f) * mk[k];    // IEEE div, matches XLA
    zmax = fmaxf(zmax, zinv[k]);
  }
  // log(EPS + prob) has only 5 possible inputs {1e-10, .25, .5, .75, 1.0}
  // (EPS+prob rounds to exactly those) -> exact constant selection, no log().
  float zmap[9];
#pragma unroll
  for (int k = 0; k < 8; ++k) {
    const int ce = (mk[k] != 0.0f) ? cntk[k] : 0;
    const float lg = (ce == 0) ? -23.025850929940457f
                   : (ce == 1) ? -1.3862943611198906f
                   : (ce == 2) ? -0.6931471805599453f
                   : (ce == 3) ? -0.2876820724517809f
                               : 0.0f;
    zmap[k] = lg + (zinv[k] - zmax) / 0.1f;          // IEEE div by GAMMA
  }
  zmap[8] = (1e-10f - zmax) / 0.1f;

  // ---- perturbed argmax over 9 entries, 4 samples (running max, first-hit tie)
  const uint32_t baseA = pix * 9u;
  float best[4];
  int bidx[4];
#pragma unroll
  for (int j = 0; j < 9; ++j) {
    const uint32_t c0 = baseA + (uint32_t)j;
    const U2 a02 = tf2x32(KA0, KA1, c0,          c0 + 2u * SA_STR);
    const U2 a13 = tf2x32(KA0, KA1, c0 + SA_STR, c0 + 3u * SA_STR);
    const float ns[4] = {bits_to_normal(a02.a), bits_to_normal(a13.a),
                         bits_to_normal(a02.b), bits_to_normal(a13.b)};
    const float z = zmap[j];
#pragma unroll
    for (int s = 0; s < 4; ++s) {
      const float v = fmaf(0.1f, ns[s], z);
      if (j == 0) { best[s] = v; bidx[s] = 0; }
      else if (v > best[s]) { best[s] = v; bidx[s] = j; }
    }
  }

  // ---- weighted color blend
  const float bgx = bg[0], bgy = bg[1], bgz = bg[2];

#if HAVE_ASYNC_LDS
  asm volatile("s_wait_asynccnt 0x0" ::: "memory");  // colors now resident in LDS
#define CKC(k, c) mycol[(k) * 3 + (c)]
#else
#define CKC(k, c) gcol[(k) * 3 + (c)]
#endif

  float r = 0.f, g = 0.f, b = 0.f;
#pragma unroll
  for (int k = 0; k < 8; ++k) {
    const float w = 0.25f * (float)((bidx[0] == k) + (bidx[1] == k) +
                                    (bidx[2] == k) + (bidx[3] == k));
    r = fmaf(w, CKC(k, 0), r);
    g = fmaf(w, CKC(k, 1), g);
    b = fmaf(w, CKC(k, 2), b);
  }
  const float wbg = 0.25f * (float)((bidx[0] == 8) + (bidx[1] == 8) +
                                    (bidx[2] == 8) + (bidx[3] == 8));
  r = fmaf(wbg, bgx, r);
  g = fmaf(wbg, bgy, g);
  b = fmaf(wbg, bgz, b);

  *(float4*)(out + (size_t)pix * 4u) = make_float4(r, g, b, 1.0f - alpha);
}

extern "C" void kernel_launch(void* const* d_in, const int* in_sizes, int n_in,
                              void* d_out, int out_size, void* d_ws,
                              size_t ws_size, hipStream_t stream) {
  (void)in_sizes; (void)n_in; (void)out_size; (void)d_ws; (void)ws_size;
  const float* colors = (const float*)d_in[0];
  const float* dists  = (const float*)d_in[1];
  const float* zbuf   = (const float*)d_in[2];
  const int*   ptf    = (const int*)d_in[3];
  const float* bg     = (const float*)d_in[4];
  float* out = (float*)d_out;

  dim3 grid(NPIX / 256u), block(256);
  rps_kernel<<<grid, block, 0, stream>>>(colors, dists, zbuf, ptf, bg, out);
}